// DFTD3_18580028522578
// MI455X (gfx1250) — compile-verified
//
#include <hip/hip_runtime.h>
#include <hip/hip_bf16.h>

typedef float v2f __attribute__((ext_vector_type(2)));
typedef float v8f __attribute__((ext_vector_type(8)));

#define BATCH   4
#define NATOM   512
#define NZTAB   95

#define F_BOHR      1.8897261258369282f
#define F_CUTOFF    15.0f
#define F_SMOOTH_ON 12.0f
#define F_K1        16.0f
#define F_K3        4.0f
#define F_A1        0.4145f
#define F_A2        4.8593f
#define F_S6        1.0f
#define F_S8        1.2177f
#define F_H2EV      27.211386245988f

// ---------------------------------------------------------------------------
// Kernel 1: pairwise r^2 via V_WMMA_F32_16X16X4_F32.
// C[M][N] = sum_k A[M][k]*B[k][N], A row i = (x,y,z,n_i), B col j = (-2x,-2y,-2z,1)
// => C = n_i - 2 x_i.x_j ; add n_j per lane afterwards.
// One wave (32 lanes) per 16x16 tile; grid = (tiles_j, tiles_i, B).
// ---------------------------------------------------------------------------
__global__ __launch_bounds__(32)
void d3_d2_wmma_kernel(const float* __restrict__ coord, float* __restrict__ d2out) {
    const int b    = blockIdx.z;
    const int ti   = blockIdx.y;
    const int tj   = blockIdx.x;
    const int lane = threadIdx.x;
    const int half = lane >> 4;    // 0: K={0,1}, 1: K={2,3}
    const int m    = lane & 15;

    const int i = ti * 16 + m;
    const int j = tj * 16 + m;

    const float* ci = coord + ((size_t)b * NATOM + i) * 3;
    const float* cj = coord + ((size_t)b * NATOM + j) * 3;
    const float xi = ci[0], yi = ci[1], zi = ci[2];
    const float xj = cj[0], yj = cj[1], zj = cj[2];
    const float ni = xi * xi + yi * yi + zi * zi;
    const float nj = xj * xj + yj * yj + zj * zj;

    // Branch-free operand packing (keeps EXEC all-ones for WMMA).
    v2f a, bv;
    a.x  = half ? zi : xi;
    a.y  = half ? ni : yi;
    bv.x = half ? (-2.0f * zj) : (-2.0f * xj);
    bv.y = half ? 1.0f         : (-2.0f * yj);

    v8f c = {};
    c = __builtin_amdgcn_wmma_f32_16x16x4_f32(
            /*neg_a=*/false, a, /*neg_b=*/false, bv,
            /*c_mod=*/(short)0, c, /*reuse_a=*/false, /*reuse_b=*/false);

    // C/D layout: VGPR p -> row M = p + 8*half ; col N = lane&15.
    float* ocol = d2out + ((size_t)b * NATOM) * NATOM + (size_t)tj * 16 + m;
#pragma unroll
    for (int p = 0; p < 8; ++p) {
        const int row = ti * 16 + p + 8 * half;
        ocol[(size_t)row * NATOM] = c[p] + nj;
    }
}

__device__ __forceinline__ float pair_d2_inline(const float* __restrict__ coord,
                                                int b, int i, int j) {
    const float* ci = coord + ((size_t)b * NATOM + i) * 3;
    const float* cj = coord + ((size_t)b * NATOM + j) * 3;
    const float dx = ci[0] - cj[0];
    const float dy = ci[1] - cj[1];
    const float dz = ci[2] - cj[2];
    return dx * dx + dy * dy + dz * dz;
}

__device__ __forceinline__ float block_reduce_256(float v) {
    __shared__ float red[256];
    const int tid = threadIdx.x;
    red[tid] = v;
    __syncthreads();
#pragma unroll
    for (int s = 128; s > 0; s >>= 1) {
        if (tid < s) red[tid] += red[tid + s];
        __syncthreads();
    }
    return red[0];
}

// ---------------------------------------------------------------------------
// Kernel 2: coordination numbers. One block per (b,i) row.
// ---------------------------------------------------------------------------
__global__ __launch_bounds__(256)
void d3_cn_kernel(const float* __restrict__ coord, const float* __restrict__ d2buf,
                  const int* __restrict__ numbers, const float* __restrict__ rcov,
                  float* __restrict__ cnout, int use_d2) {
    const int i   = blockIdx.x;
    const int b   = blockIdx.y;
    const int tid = threadIdx.x;

    const int   zn_i = numbers[b * NATOM + i];
    const float rci  = rcov[zn_i];

    float acc = 0.0f;
    for (int j = tid; j < NATOM; j += 256) {
        float d2 = use_d2 ? d2buf[((size_t)b * NATOM + i) * NATOM + j]
                          : pair_d2_inline(coord, b, i, j);
        const float r_ang = sqrtf(fmaxf(d2, 0.0f));
        if ((j != i) && (r_ang <= F_CUTOFF)) {
            const float rcj = rcov[numbers[b * NATOM + j]];
            const float r   = r_ang * F_BOHR;
            const float arg = -F_K1 * ((rci + rcj) / r - 1.0f);
            acc += 1.0f / (1.0f + __expf(arg));
        }
    }
    const float total = block_reduce_256(acc);
    if (tid == 0) cnout[b * NATOM + i] = total;
}

// ---------------------------------------------------------------------------
// Kernel 3: per-pair dispersion energy, summed over j for each row (b,i).
// ---------------------------------------------------------------------------
__global__ __launch_bounds__(256)
void d3_energy_kernel(const float* __restrict__ coord, const float* __restrict__ d2buf,
                      const int* __restrict__ numbers, const float* __restrict__ r4r2,
                      const float* __restrict__ c6ab, const float* __restrict__ cn_ref,
                      const float* __restrict__ cnbuf, float* __restrict__ rowE,
                      int use_d2) {
    const int i   = blockIdx.x;
    const int b   = blockIdx.y;
    const int tid = threadIdx.x;

    const int   zn_i = numbers[b * NATOM + i];
    const float cn_i = cnbuf[b * NATOM + i];
    const float rr_i = r4r2[zn_i];

    float acc = 0.0f;
    for (int j = tid; j < NATOM; j += 256) {
        float d2 = use_d2 ? d2buf[((size_t)b * NATOM + i) * NATOM + j]
                          : pair_d2_inline(coord, b, i, j);
        d2 = fmaxf(d2, 0.0f);
        const float r_ang = sqrtf(d2);
        if ((j != i) && (r_ang <= F_CUTOFF)) {
            const int   zn_j = numbers[b * NATOM + j];
            const float cn_j = cnbuf[b * NATOM + j];

            const float* c6p  = c6ab   + ((size_t)zn_i * NZTAB + zn_j) * 25;
            const float* cnip = cn_ref + ((size_t)zn_i * NZTAB + zn_j) * 25;
            const float* cnjp = cn_ref + ((size_t)zn_j * NZTAB + zn_i) * 25;

            float wsum = 0.0f, c6w = 0.0f;
#pragma unroll
            for (int ai = 0; ai < 5; ++ai) {
#pragma unroll
                for (int aj = 0; aj < 5; ++aj) {
                    const float c6r = c6p[ai * 5 + aj];
                    const float di  = cn_i - cnip[ai * 5 + aj];
                    const float dj  = cn_j - cnjp[aj * 5 + ai];  // transposed gather
                    const float w   = (c6r > 0.0f)
                                        ? __expf(-F_K3 * (di * di + dj * dj))
                                        : 0.0f;
                    wsum += w;
                    c6w  += c6r * w;
                }
            }
            const float c6   = c6w / (wsum + 1e-20f);
            const float rrij = rr_i * r4r2[zn_j];
            const float c8   = 3.0f * c6 * rrij;
            const float r0   = sqrtf(3.0f * rrij);
            const float f    = F_A1 * r0 + F_A2;

            const float rb2 = d2 * (F_BOHR * F_BOHR);
            const float r6  = rb2 * rb2 * rb2;
            const float r8  = r6 * rb2;
            const float f2  = f * f;
            const float f6  = f2 * f2 * f2;
            const float f8  = f6 * f2;

            float e = F_S6 * c6 / (r6 + f6) + F_S8 * c8 / (r8 + f8);

            float x = (r_ang - F_SMOOTH_ON) * (1.0f / (F_CUTOFF - F_SMOOTH_ON));
            x = fminf(fmaxf(x, 0.0f), 1.0f);
            const float sw = 1.0f - x * x * (3.0f - 2.0f * x);
            acc += e * sw;
        }
    }
    const float total = block_reduce_256(acc);
    if (tid == 0) rowE[b * NATOM + i] = total;
}

// ---------------------------------------------------------------------------
// Kernel 4: final per-batch reduction (4 blocks, deterministic tree).
// ---------------------------------------------------------------------------
__global__ __launch_bounds__(256)
void d3_reduce_kernel(const float* __restrict__ rowE, float* __restrict__ out) {
    const int b   = blockIdx.x;
    const int tid = threadIdx.x;
    float acc = rowE[b * NATOM + tid] + rowE[b * NATOM + tid + 256];
    const float total = block_reduce_256(acc);
    if (tid == 0) out[b] = -0.5f * F_H2EV * total;
}

extern "C" void kernel_launch(void* const* d_in, const int* in_sizes, int n_in,
                              void* d_out, int out_size, void* d_ws, size_t ws_size,
                              hipStream_t stream) {
    const float* coord   = (const float*)d_in[0];   // (4,512,3) f32
    const int*   numbers = (const int*)d_in[1];     // (4,512) int32
    const float* rcov    = (const float*)d_in[2];   // (95,) f32
    const float* r4r2    = (const float*)d_in[3];   // (95,) f32
    const float* c6ab    = (const float*)d_in[4];   // (95,95,5,5) f32
    const float* cn_ref  = (const float*)d_in[5];   // (95,95,5,5) f32
    float*       out     = (float*)d_out;           // (4,) f32
    (void)in_sizes; (void)n_in; (void)out_size;

    const size_t d2_bytes    = (size_t)BATCH * NATOM * NATOM * sizeof(float); // 4 MB
    const size_t small_elems = (size_t)BATCH * NATOM;                         // 2048
    const int use_d2 = (ws_size >= d2_bytes + 2 * small_elems * sizeof(float)) ? 1 : 0;

    float* d2buf = (float*)d_ws;
    float* cnbuf = use_d2 ? (float*)((char*)d_ws + d2_bytes) : (float*)d_ws;
    float* rowE  = cnbuf + small_elems;

    if (use_d2) {
        dim3 grid(NATOM / 16, NATOM / 16, BATCH); // 32 x 32 x 4 tiles
        d3_d2_wmma_kernel<<<grid, 32, 0, stream>>>(coord, d2buf);
    }

    {
        dim3 grid(NATOM, BATCH);
        d3_cn_kernel<<<grid, 256, 0, stream>>>(coord, d2buf, numbers, rcov, cnbuf, use_d2);
        d3_energy_kernel<<<grid, 256, 0, stream>>>(coord, d2buf, numbers, r4r2,
                                                   c6ab, cn_ref, cnbuf, rowE, use_d2);
    }

    d3_reduce_kernel<<<BATCH, 256, 0, stream>>>(rowE, out);
}